// PhysicsLoss_54537494725163
// MI455X (gfx1250) — compile-verified
//
#include <hip/hip_runtime.h>
#include <hip/hip_bf16.h>

// ---------------------------------------------------------------------------
// CDNA5 (gfx1250) implementation of the DeepONet elasticity physics loss.
// Heavy math = analytic Hessian propagation through the trunk MLP, expressed
// as large f32 GEMMs executed with V_WMMA_F32_16X16X4_F32.
// Trunk GEMMs stage B tiles in LDS with a row-pair-interleaved layout so each
// WMMA B fragment is one conflict-free ds_load_b64 into an aligned VGPR pair.
// ---------------------------------------------------------------------------

typedef __attribute__((ext_vector_type(2))) float v2f;
typedef __attribute__((ext_vector_type(8))) float v8f;

static __device__ __forceinline__ v8f wmma_f32_4(v2f a, v2f b, v8f c) {
  // 8 args: (neg_a, A, neg_b, B, c_mod, C, reuse_a, reuse_b)
  return __builtin_amdgcn_wmma_f32_16x16x4_f32(false, a, false, b, (short)0, c,
                                               false, false);
}

// problem constants
constexpr float NU_    = 0.3f;
constexpr float PREF_  = 1.0f / (1.0f - NU_ * NU_);
constexpr int   BB     = 64;      // batch
constexpr int   N_INT  = 20000;
constexpr int   N_DIR  = 512;
constexpr int   N_FOR  = 101;
constexpr int   NPTS   = 20672;   // 20613 padded to multiple of 64
constexpr int   NSTRM  = 6;       // [val, Jx, Jy, Hxx, Hxy, Hyy]
constexpr int   NFCOL  = NSTRM * NPTS; // 124032 columns, multiple of 64

// ---------------------------------------------------------------------------
// Generic f32 WMMA GEMM: C(MxN) = act( A(MxK) * B(KxN) + bias[n] )
// A row-major (lda). B row-major (ldb) if !BT, else B[n*ldb + k].
// One wave -> 16(M) x 16*NT(N); block = 8 waves stacked along M.
// Used for the small branch linear layers.
// ---------------------------------------------------------------------------
template <bool BT, int NT>
__global__ void gemm_wmma_f32(const float* __restrict__ A,
                              const float* __restrict__ B,
                              float* __restrict__ C,
                              const float* __restrict__ bias,
                              int M, int N, int K,
                              int lda, int ldb, int ldc, int doTanh) {
  const int lane = threadIdx.x & 31;
  const int wave = threadIdx.x >> 5;
  const int half = lane >> 4;
  const int l16  = lane & 15;
  const int m0   = (blockIdx.y * (int)(blockDim.x >> 5) + wave) * 16;
  const int n0   = blockIdx.x * (16 * NT);
  if (m0 >= M) return;  // wave-uniform

  v8f acc[NT];
#pragma unroll
  for (int t = 0; t < NT; ++t)
#pragma unroll
    for (int r = 0; r < 8; ++r) acc[t][r] = 0.0f;

  const float* Arow = A + (size_t)(m0 + l16) * lda;
  for (int k = 0; k < K; k += 4) {
    v2f a;
    a.x = Arow[k + 2 * half + 0];
    a.y = Arow[k + 2 * half + 1];
#pragma unroll
    for (int t = 0; t < NT; ++t) {
      const int n = n0 + t * 16 + l16;
      v2f b;
      if (BT) {
        const float* Bc = B + (size_t)n * ldb;
        b.x = Bc[k + 2 * half + 0];
        b.y = Bc[k + 2 * half + 1];
      } else {
        b.x = B[(size_t)(k + 2 * half + 0) * ldb + n];
        b.y = B[(size_t)(k + 2 * half + 1) * ldb + n];
      }
      acc[t] = wmma_f32_4(a, b, acc[t]);
    }
  }

#pragma unroll
  for (int t = 0; t < NT; ++t) {
#pragma unroll
    for (int r = 0; r < 8; ++r) {
      const int m = m0 + r + 8 * half;
      const int n = n0 + t * 16 + l16;
      float v = acc[t][r];
      if (bias) v += bias[n];
      if (doTanh) v = tanhf(v);
      C[(size_t)m * ldc + n] = v;
    }
  }
}

// ---------------------------------------------------------------------------
// Trunk GEMM: C(256 x NFCOL) = A(256x256) @ B(256 x NFCOL), 64 cols per block.
// 512 threads = 16 waves -> all 256 output rows in one block, so each B tile
// is pulled from HBM exactly once and served to all waves from LDS.
// LDS layout interleaves row pairs: element (k,n) -> bs[(k/2)*160 + 2n + (k&1)]
// so a B fragment {B[k][n], B[k+1][n]} is ONE aligned ds_load_b64.
// Row-pair stride 160 floats: half-wave offset = 160 mod 64 = 32 banks, each
// 16-lane half spans 32 banks (stride-2, b64) -> disjoint windows, no conflict.
// ---------------------------------------------------------------------------
__global__ void gemm_trunk_lds(const float* __restrict__ A,
                               const float* __restrict__ B,
                               float* __restrict__ C) {
  constexpr int K    = 256;
  constexpr int LDSS = 160;            // floats per k-pair row (128 + 32 pad)
  extern __shared__ float bs[];        // 128 * 160 * 4 = 81920 bytes

  const int tid  = threadIdx.x;   // 0..511
  const int lane = tid & 31;
  const int wave = tid >> 5;      // 0..15
  const int half = lane >> 4;
  const int l16  = lane & 15;
  const int n0   = blockIdx.x * 64;

  // cooperative load of B tile (256 rows x 64 cols), pair-interleaved in LDS
  {
    const float* Bs = B + n0;
#pragma unroll
    for (int r = 0; r < 4; ++r) {
      const int q  = r * 512 + tid;   // 0..2047 tasks (128 kpairs x 16 groups)
      const int kp = q >> 4;
      const int c4 = (q & 15) << 2;
      const float4 v0 =
          *reinterpret_cast<const float4*>(Bs + (size_t)(2 * kp + 0) * NFCOL + c4);
      const float4 v1 =
          *reinterpret_cast<const float4*>(Bs + (size_t)(2 * kp + 1) * NFCOL + c4);
      float2* d = reinterpret_cast<float2*>(&bs[kp * LDSS + c4 * 2]);
      d[0] = make_float2(v0.x, v1.x);
      d[1] = make_float2(v0.y, v1.y);
      d[2] = make_float2(v0.z, v1.z);
      d[3] = make_float2(v0.w, v1.w);
    }
  }
  __syncthreads();

  v8f acc[4];
#pragma unroll
  for (int t = 0; t < 4; ++t)
#pragma unroll
    for (int r = 0; r < 8; ++r) acc[t][r] = 0.0f;

  const int m0 = wave * 16;
  const float* Arow = A + (size_t)(m0 + l16) * K;
#pragma unroll 4
  for (int k = 0; k < K; k += 4) {
    v2f a;
    a.x = Arow[k + 2 * half + 0];
    a.y = Arow[k + 2 * half + 1];
    const int kp = (k >> 1) + half;   // k-pair index for this half-wave
#pragma unroll
    for (int t = 0; t < 4; ++t) {
      const float2 bb = *reinterpret_cast<const float2*>(
          &bs[kp * LDSS + (t * 16 + l16) * 2]);
      v2f b;
      b.x = bb.x;
      b.y = bb.y;
      acc[t] = wmma_f32_4(a, b, acc[t]);
    }
  }

#pragma unroll
  for (int t = 0; t < 4; ++t) {
#pragma unroll
    for (int r = 0; r < 8; ++r) {
      const int m = m0 + r + 8 * half;
      const int n = n0 + t * 16 + l16;
      C[(size_t)m * NFCOL + n] = acc[t][r];
    }
  }
}

// ---------------------------------------------------------------------------
// GEMM + square-accumulate: accum += scale * sum((A(64xK) * B(KxN))^2)
// M fixed = 64 (4 M-tiles, waves 0-3 / 4-7 split 64 N-cols). N multiple of 64.
// ---------------------------------------------------------------------------
__global__ void gemm_sq_wmma_f32(const float* __restrict__ A,
                                 const float* __restrict__ B,
                                 float* __restrict__ accum,
                                 int K, int lda, int ldb, float scale) {
  const int lane = threadIdx.x & 31;
  const int wave = threadIdx.x >> 5;
  const int half = lane >> 4;
  const int l16  = lane & 15;
  const int m0   = (wave & 3) * 16;
  const int n0   = blockIdx.x * 64 + (wave >> 2) * 32;

  v8f acc[2];
#pragma unroll
  for (int t = 0; t < 2; ++t)
#pragma unroll
    for (int r = 0; r < 8; ++r) acc[t][r] = 0.0f;

  const float* Arow = A + (size_t)(m0 + l16) * lda;
  for (int k = 0; k < K; k += 4) {
    v2f a;
    a.x = Arow[k + 2 * half + 0];
    a.y = Arow[k + 2 * half + 1];
#pragma unroll
    for (int t = 0; t < 2; ++t) {
      const int n = n0 + t * 16 + l16;
      v2f b;
      b.x = B[(size_t)(k + 2 * half + 0) * ldb + n];
      b.y = B[(size_t)(k + 2 * half + 1) * ldb + n];
      acc[t] = wmma_f32_4(a, b, acc[t]);
    }
  }

  float local = 0.0f;
#pragma unroll
  for (int t = 0; t < 2; ++t)
#pragma unroll
    for (int r = 0; r < 8; ++r) local += acc[t][r] * acc[t][r];

  __shared__ float red[256];
  red[threadIdx.x] = local;
  __syncthreads();
  for (int s = 128; s > 0; s >>= 1) {
    if ((int)threadIdx.x < s) red[threadIdx.x] += red[threadIdx.x + s];
    __syncthreads();
  }
  if (threadIdx.x == 0) atomicAdd(accum, red[0] * scale);
}

// ---------------------------------------------------------------------------
// Branch convolutions
// ---------------------------------------------------------------------------
__global__ void branch_conv1(const float* __restrict__ f,
                             const float* __restrict__ w,
                             const float* __restrict__ b,
                             float* __restrict__ out) {
  const int idx = blockIdx.x * blockDim.x + threadIdx.x; // 64*64*101
  if (idx >= BB * 64 * 101) return;
  const int pos = idx % 101;
  const int oc  = (idx / 101) & 63;
  const int bi  = idx / (101 * 64);
  float s = b[oc];
#pragma unroll
  for (int k = 0; k < 5; ++k) {
    const int p = pos + k - 2;
    if (p < 0 || p >= 101) continue;
    const float x0 = f[bi * 202 + p];
    const float x1 = f[bi * 202 + 101 + p];
    const float x2 = (float)p * 0.01f;  // linspace(0,1,101)
    s += w[(oc * 3 + 0) * 5 + k] * x0 +
         w[(oc * 3 + 1) * 5 + k] * x1 +
         w[(oc * 3 + 2) * 5 + k] * x2;
  }
  out[idx] = tanhf(s);
}

__global__ void branch_conv64(const float* __restrict__ hin,
                              const float* __restrict__ w,
                              const float* __restrict__ b,
                              float* __restrict__ out) {
  const int idx = blockIdx.x * blockDim.x + threadIdx.x;
  if (idx >= BB * 64 * 101) return;
  const int pos = idx % 101;
  const int oc  = (idx / 101) & 63;
  const int bi  = idx / (101 * 64);
  float s = b[oc];
  for (int ic = 0; ic < 64; ++ic) {
    const float* row = hin + (size_t)(bi * 64 + ic) * 101;
    const float* wr  = w + (size_t)(oc * 64 + ic) * 5;
#pragma unroll
    for (int k = 0; k < 5; ++k) {
      const int p = pos + k - 2;
      if (p >= 0 && p < 101) s += row[p] * wr[k];
    }
  }
  out[idx] = tanhf(s);
}

// ---------------------------------------------------------------------------
// Trunk layer 1 + analytic derivative seeds (feature-major: row=unit, col=pt)
// streams: 0=h, 1=dh/dx, 2=dh/dy, 3=d2/dxx, 4=d2/dxy, 5=d2/dyy
// ---------------------------------------------------------------------------
__global__ void trunk_layer1(const float* __restrict__ xy_i,
                             const float* __restrict__ xy_d,
                             const float* __restrict__ xy_f,
                             const float* __restrict__ tw1,
                             const float* __restrict__ tb1,
                             float* __restrict__ feat) {
  const size_t idx = (size_t)blockIdx.x * blockDim.x + threadIdx.x;
  if (idx >= (size_t)256 * NPTS) return;
  const int n = (int)(idx % NPTS);
  const int j = (int)(idx / NPTS);
  float x = 0.0f, y = 0.0f;
  if (n < N_INT)                { x = xy_i[2 * n];             y = xy_i[2 * n + 1]; }
  else if (n < N_INT + N_DIR)   { const int m = n - N_INT;     x = xy_d[2 * m]; y = xy_d[2 * m + 1]; }
  else if (n < N_INT + N_DIR + N_FOR) { const int m = n - N_INT - N_DIR; x = xy_f[2 * m]; y = xy_f[2 * m + 1]; }
  const float w0 = tw1[2 * j], w1 = tw1[2 * j + 1];
  const float t  = tanhf(w0 * x + w1 * y + tb1[j]);
  const float a  = 1.0f - t * t;     // tanh'
  const float s  = -2.0f * t * a;    // tanh''
  float* p = feat + (size_t)j * NFCOL + n;
  p[0 * NPTS] = t;
  p[1 * NPTS] = a * w0;
  p[2 * NPTS] = a * w1;
  p[3 * NPTS] = s * w0 * w0;
  p[4 * NPTS] = s * w0 * w1;
  p[5 * NPTS] = s * w1 * w1;
}

// chain rule through tanh for layer 2 (in-place safe: 1 thread owns 6 slots)
__global__ void trunk_mid(const float* __restrict__ g,
                          const float* __restrict__ tb2,
                          float* __restrict__ feat2) {
  const size_t idx = (size_t)blockIdx.x * blockDim.x + threadIdx.x;
  if (idx >= (size_t)256 * NPTS) return;
  const int n = (int)(idx % NPTS);
  const int j = (int)(idx / NPTS);
  const size_t base = (size_t)j * NFCOL + n;
  const float z   = g[base + 0 * NPTS] + tb2[j];
  const float dzx = g[base + 1 * NPTS];
  const float dzy = g[base + 2 * NPTS];
  const float hxx = g[base + 3 * NPTS];
  const float hxy = g[base + 4 * NPTS];
  const float hyy = g[base + 5 * NPTS];
  const float t = tanhf(z);
  const float a = 1.0f - t * t;
  const float s = -2.0f * t * a;
  feat2[base + 0 * NPTS] = t;
  feat2[base + 1 * NPTS] = a * dzx;
  feat2[base + 2 * NPTS] = a * dzy;
  feat2[base + 3 * NPTS] = s * dzx * dzx + a * hxx;
  feat2[base + 4 * NPTS] = s * dzx * dzy + a * hxy;
  feat2[base + 5 * NPTS] = s * dzy * dzy + a * hyy;
}

// Build elasticity residual operators Ru/Rv (256 x 2*N_INT) from Hessians.
__global__ void build_R(const float* __restrict__ out3, float* __restrict__ R) {
  const size_t idx = (size_t)blockIdx.x * blockDim.x + threadIdx.x;
  if (idx >= (size_t)256 * N_INT) return;
  const int n = (int)(idx % N_INT);
  const int k = (int)(idx / N_INT);
  const size_t c = (size_t)k * NFCOL + n;
  const float hxx = out3[c + 3 * NPTS];
  const float hxy = out3[c + 4 * NPTS];
  const float hyy = out3[c + 5 * NPTS];
  const float c1 = 0.5f * (1.0f - NU_);
  const float c2 = 0.5f * (1.0f + NU_);
  float ru, rv;
  if (k < 128) { ru = PREF_ * (hxx + c1 * hyy); rv = PREF_ * c2 * hxy; }
  else         { ru = PREF_ * c2 * hxy;         rv = PREF_ * (hyy + c1 * hxx); }
  R[(size_t)k * (2 * N_INT) + n]         = ru;
  R[(size_t)k * (2 * N_INT) + N_INT + n] = rv;
}

// Dirichlet loss: uv[b,n,c] = sum_p bc[b,c*128+p] * (val[c*128+p,n] + tb3)
__global__ void dir_loss(const float* __restrict__ bc,
                         const float* __restrict__ out3,
                         const float* __restrict__ tb3,
                         float* __restrict__ accum) {
  const int idx = blockIdx.x * blockDim.x + threadIdx.x; // 64*512
  float val = 0.0f;
  if (idx < BB * N_DIR) {
    const int b = idx & 63;
    const int n = idx >> 6;
    const int col = N_INT + n;
    float s0 = 0.0f, s1 = 0.0f;
    for (int p = 0; p < 128; ++p) {
      const float t0 = out3[(size_t)p * NFCOL + col] + tb3[p];
      const float t1 = out3[(size_t)(128 + p) * NFCOL + col] + tb3[128 + p];
      s0 += bc[b * 256 + p] * t0;
      s1 += bc[b * 256 + 128 + p] * t1;
    }
    val = (s0 * s0 + s1 * s1) * (1.0f / (float)(BB * N_DIR)); // W_DIR = 1
  }
  __shared__ float red[256];
  red[threadIdx.x] = val;
  __syncthreads();
  for (int s = 128; s > 0; s >>= 1) {
    if ((int)threadIdx.x < s) red[threadIdx.x] += red[threadIdx.x + s];
    __syncthreads();
  }
  if (threadIdx.x == 0) atomicAdd(accum, red[0]);
}

// Forcing loss at 101 boundary points using Jacobian streams.
__global__ void force_loss(const float* __restrict__ bc,
                           const float* __restrict__ out3,
                           const float* __restrict__ f,
                           float* __restrict__ accum) {
  const int idx = blockIdx.x * blockDim.x + threadIdx.x; // 64*101
  float val = 0.0f;
  if (idx < BB * N_FOR) {
    const int b = idx % BB;
    const int n = idx / BB;
    const int col = N_INT + N_DIR + n;
    float ux = 0.0f, uy = 0.0f, vx = 0.0f, vy = 0.0f;
    for (int p = 0; p < 128; ++p) {
      const float jx0 = out3[(size_t)p * NFCOL + 1 * NPTS + col];
      const float jy0 = out3[(size_t)p * NFCOL + 2 * NPTS + col];
      const float jx1 = out3[(size_t)(128 + p) * NFCOL + 1 * NPTS + col];
      const float jy1 = out3[(size_t)(128 + p) * NFCOL + 2 * NPTS + col];
      const float b0 = bc[b * 256 + p];
      const float b1 = bc[b * 256 + 128 + p];
      ux += b0 * jx0; uy += b0 * jy0;
      vx += b1 * jx1; vy += b1 * jy1;
    }
    const float gx = f[b * 202 + n];
    const float gy = f[b * 202 + 101 + n];
    const float lu = PREF_ * (ux + NU_ * vy) + gx;
    const float lv = PREF_ * (1.0f - NU_) * 0.5f * (uy + vx) + gy;
    val = (lu * lu + lv * lv) * (1.0f / (float)(BB * N_FOR)); // W_FORCE = 1
  }
  __shared__ float red[256];
  red[threadIdx.x] = val;
  __syncthreads();
  for (int s = 128; s > 0; s >>= 1) {
    if ((int)threadIdx.x < s) red[threadIdx.x] += red[threadIdx.x + s];
    __syncthreads();
  }
  if (threadIdx.x == 0) atomicAdd(accum, red[0]);
}

__global__ void zero_scalar(float* p) { if (threadIdx.x == 0) p[0] = 0.0f; }

// ---------------------------------------------------------------------------
extern "C" void kernel_launch(void* const* d_in, const int* in_sizes, int n_in,
                              void* d_out, int out_size, void* d_ws, size_t ws_size,
                              hipStream_t stream) {
  const float* f_batch = (const float*)d_in[0];
  const float* xy_int  = (const float*)d_in[1];
  const float* xy_dir  = (const float*)d_in[2];
  const float* xy_for  = (const float*)d_in[3];
  const float* cw1 = (const float*)d_in[4];
  const float* cb1 = (const float*)d_in[5];
  const float* cw2 = (const float*)d_in[6];
  const float* cb2 = (const float*)d_in[7];
  const float* cw3 = (const float*)d_in[8];
  const float* cb3 = (const float*)d_in[9];
  const float* lw1 = (const float*)d_in[10];
  const float* lb1 = (const float*)d_in[11];
  const float* lw2 = (const float*)d_in[12];
  const float* lb2 = (const float*)d_in[13];
  const float* lw3 = (const float*)d_in[14];
  const float* lb3 = (const float*)d_in[15];
  const float* tw1 = (const float*)d_in[16];
  const float* tb1 = (const float*)d_in[17];
  const float* tw2 = (const float*)d_in[18];
  const float* tb2 = (const float*)d_in[19];
  const float* tw3 = (const float*)d_in[20];
  const float* tb3 = (const float*)d_in[21];
  float* out = (float*)d_out;

  // workspace layout (floats)
  float* ws = (float*)d_ws;
  size_t o = 0;
  float* h1  = ws + o; o += (size_t)BB * 64 * 101;
  float* h2  = ws + o; o += (size_t)BB * 64 * 101;
  float* h3  = ws + o; o += (size_t)BB * 64 * 101;
  float* b1  = ws + o; o += (size_t)BB * 256;
  float* b2  = ws + o; o += (size_t)BB * 256;
  float* bcm = ws + o; o += (size_t)BB * 256;   // branch coefficients (64x256)
  float* big1 = ws + o; o += (size_t)256 * NFCOL;  // feat1 -> out3
  float* big2 = ws + o; o += (size_t)256 * NFCOL;  // z-streams -> feat2
  float* Rmat = ws + o; o += (size_t)256 * (2 * N_INT);
  (void)ws_size; (void)n_in; (void)in_sizes; (void)out_size;

  zero_scalar<<<1, 64, 0, stream>>>(out);

  // ---- branch net ----
  {
    const int total = BB * 64 * 101;
    const int blocks = (total + 255) / 256;
    branch_conv1<<<blocks, 256, 0, stream>>>(f_batch, cw1, cb1, h1);
    branch_conv64<<<blocks, 256, 0, stream>>>(h1, cw2, cb2, h2);
    branch_conv64<<<blocks, 256, 0, stream>>>(h2, cw3, cb3, h3);
  }
  // linear layers via WMMA (B transposed: B[n*ldb+k] = W[n,k])
  gemm_wmma_f32<true, 4><<<dim3(256 / 64, 1), 256, 0, stream>>>(
      h3, lw1, b1, lb1, BB, 256, 6464, 6464, 6464, 256, 1);
  gemm_wmma_f32<true, 4><<<dim3(256 / 64, 1), 256, 0, stream>>>(
      b1, lw2, b2, lb2, BB, 256, 256, 256, 256, 256, 1);
  gemm_wmma_f32<true, 4><<<dim3(256 / 64, 1), 256, 0, stream>>>(
      b2, lw3, bcm, lb3, BB, 256, 256, 256, 256, 256, 0);

  // ---- trunk with analytic value/Jacobian/Hessian propagation ----
  {
    const size_t total = (size_t)256 * NPTS;
    const int blocks = (int)((total + 255) / 256);
    const size_t ldsBytes = 128 * 160 * sizeof(float);  // 81920 B
    trunk_layer1<<<blocks, 256, 0, stream>>>(xy_int, xy_dir, xy_for, tw1, tb1, big1);
    // layer 2: z-streams = tw2 @ feat1   (256 x NFCOL)
    gemm_trunk_lds<<<NFCOL / 64, 512, ldsBytes, stream>>>(tw2, big1, big2);
    trunk_mid<<<blocks, 256, 0, stream>>>(big2, tb2, big2);  // in place
    // layer 3: out-streams = tw3 @ feat2
    gemm_trunk_lds<<<NFCOL / 64, 512, ldsBytes, stream>>>(tw3, big2, big1);
  }

  // ---- interior PDE residual loss via fused GEMM + square-sum ----
  {
    const size_t total = (size_t)256 * N_INT;
    build_R<<<(int)((total + 255) / 256), 256, 0, stream>>>(big1, Rmat);
    const float scale = 1.0f / (float)(BB * N_INT);  // W_INT = 1
    gemm_sq_wmma_f32<<<(2 * N_INT) / 64, 256, 0, stream>>>(
        bcm, Rmat, out, 256, 256, 2 * N_INT, scale);
  }

  // ---- dirichlet + forcing losses ----
  dir_loss<<<(BB * N_DIR + 255) / 256, 256, 0, stream>>>(bcm, big1, tb3, out);
  force_loss<<<(BB * N_FOR + 255) / 256, 256, 0, stream>>>(bcm, big1, f_batch, out);
}